// EdgeConv_81544249082045
// MI455X (gfx1250) — compile-verified
//
#include <hip/hip_runtime.h>

#define B_ 8
#define C_ 64
#define N_ 4096
#define K_ 20
#define O_ 64
#define EPS_ 1e-5f
#define SLOPE_ 0.2f

typedef __attribute__((ext_vector_type(2))) float v2f;
typedef __attribute__((ext_vector_type(8))) float v8f;

#if defined(__HIP_DEVICE_COMPILE__)
#if !__has_builtin(__builtin_amdgcn_wmma_f32_16x16x4_f32)
#error "missing __builtin_amdgcn_wmma_f32_16x16x4_f32 - report exact name/signature"
#endif
#endif

__device__ __forceinline__ v8f wmma4(v2f a, v2f b, v8f c) {
#if defined(__HIP_DEVICE_COMPILE__)
  // D(16x16,f32) = A(16x4,f32) x B(4x16,f32) + C
  return __builtin_amdgcn_wmma_f32_16x16x4_f32(false, a, false, b, (short)0, c,
                                               false, false);
#else
  (void)a; (void)b;
  return c;
#endif
}

__device__ __forceinline__ void wait_async0() {
#if defined(__HIP_DEVICE_COMPILE__)
  asm volatile("s_wait_asynccnt 0x0" ::: "memory");
#endif
}

// ---------------- Phase 0: xx[b][n] = sum_c x[b][c][n]^2 ----------------
__global__ __launch_bounds__(256) void sumsq_kernel(const float* __restrict__ x,
                                                    float* __restrict__ xx) {
  int i = blockIdx.x * 256 + threadIdx.x;   // over B*N
  int b = i >> 12;                          // N_ == 4096
  int n = i & (N_ - 1);
  const float* xb = x + (size_t)b * C_ * N_;
  float s = 0.f;
#pragma unroll
  for (int c = 0; c < C_; ++c) {
    float v = xb[(size_t)c * N_ + n];
    s = fmaf(v, v, s);
  }
  xx[i] = s;
}

// ---------------- Phase 1: fused pairwise-distance + top-K ----------------
#define QT 64    // queries per workgroup (16 per wave, 4 waves)
#define CT 128   // candidate tile staged in LDS
#define XP 144   // LDS pitch for candidate tile (bank-conflict padding)
#define PDP 17   // pitch of the transpose buffer

// Async DMA of x[b][0:64][m0:m0+128] into an LDS tile (row pitch XP floats).
// Each wave issues its rows (r == wave mod 4), one b128 per lane per row.
// Tracked on ASYNCcnt; caller does wait_async0() + __syncthreads().
__device__ __forceinline__ void async_stage_tile(const float* xb, int m0,
                                                 float* xsbuf, int wave,
                                                 int lane) {
#if defined(__HIP_DEVICE_COMPILE__)
  unsigned ldsbase = (unsigned)(unsigned long long)xsbuf;  // low 32 = LDS offset
#pragma unroll 1
  for (int r = wave; r < C_; r += 4) {
    unsigned laddr = ldsbase + (unsigned)(r * XP + lane * 4) * 4u;
    unsigned goff = (unsigned)(r * N_ + m0 + lane * 4) * 4u;
    asm volatile("global_load_async_to_lds_b128 %0, %1, %2"
                 :
                 : "v"(laddr), "v"(goff), "s"(xb)
                 : "memory");
  }
#else
  (void)xb; (void)m0; (void)xsbuf; (void)wave; (void)lane;
#endif
}

__global__ __launch_bounds__(128) void knn_kernel(const float* __restrict__ x,
                                                  const float* __restrict__ xx,
                                                  int* __restrict__ idxOut) {
  __shared__ float xs[2][C_ * XP];         // double-buffered candidate tile
  __shared__ float xxs[2][CT];
  __shared__ float pdt[4 * 16 * PDP];      // per-wave key transpose buffer
  __shared__ float mv[4 * 16 * 2 * K_];    // merge values
  __shared__ int   mi[4 * 16 * 2 * K_];    // merge indices

  const int tid = threadIdx.x;
  const int wave = tid >> 5;
  const int lane = tid & 31;
  const int lh = lane & 15;
  const int hif = (lane >> 4) & 1;
  const int b = blockIdx.y;
  const int q0 = blockIdx.x * QT + wave * 16;

  const float* xb = x + (size_t)b * C_ * N_;
  const float* xxb = xx + (size_t)b * N_;

  // A fragments: 16 queries x 64 channels, resident for the whole sweep.
  // A(16x4) layout: lanes 0-15 -> K={0,1}, lanes 16-31 -> K={2,3}.
  v2f afrag[16];
#pragma unroll
  for (int kc = 0; kc < 16; ++kc) {
    int c0 = 4 * kc + 2 * hif;
    afrag[kc][0] = xb[(size_t)c0 * N_ + q0 + lh];
    afrag[kc][1] = xb[(size_t)(c0 + 1) * N_ + q0 + lh];
  }

  // Per-lane private top-K (register resident; all indices static).
  float tv[K_];
  int ti[K_];
#pragma unroll
  for (int j = 0; j < K_; ++j) { tv[j] = -3.0e38f; ti[j] = 0; }

  float* pw = pdt + wave * 16 * PDP;
  const int qrow = lane >> 1;   // query this lane selects for
  const int half = lane & 1;    // which interleaved half of candidates

  // prologue: kick off DMA of tile 0
  async_stage_tile(xb, 0, &xs[0][0], wave, lane);
  if (tid < CT) xxs[0][tid] = xxb[tid];

  const int NT = N_ / CT;
  for (int it = 0; it < NT; ++it) {
    const int m0 = it * CT;
    const int cur = it & 1;
    wait_async0();        // my async writes for xs[cur] are done
    __syncthreads();      // everyone's are done and visible

    if (it + 1 < NT) {    // overlap DMA of next tile with this tile's WMMAs
      async_stage_tile(xb, m0 + CT, &xs[cur ^ 1][0], wave, lane);
      if (tid < CT) xxs[cur ^ 1][tid] = xxb[m0 + CT + tid];
    }

    const float* xsc = &xs[cur][0];
    const float* xxc = &xxs[cur][0];

#pragma unroll 1
    for (int sub = 0; sub < CT / 16; ++sub) {
      v8f acc = {0.f, 0.f, 0.f, 0.f, 0.f, 0.f, 0.f, 0.f};
      const int mloc = sub * 16 + lh;
#pragma unroll
      for (int kc = 0; kc < 16; ++kc) {
        int c0 = 4 * kc + 2 * hif;
        v2f bf;
        bf[0] = xsc[c0 * XP + mloc];
        bf[1] = xsc[(c0 + 1) * XP + mloc];
        acc = wmma4(afrag[kc], bf, acc);
      }
      // ranking key: 2*inner - xx[m]  (the -xx[q] row constant drops out)
      float xxv = xxc[mloc];
#pragma unroll
      for (int v = 0; v < 8; ++v) {
        int M = v + 8 * hif;                // C/D layout: VGPR v -> row M
        pw[M * PDP + lh] = 2.0f * acc[v] - xxv;
      }
      // transpose via LDS (same-wave DS ops are in order), then insert
#pragma unroll 1
      for (int j2 = 0; j2 < 8; ++j2) {
        int cl = half + 2 * j2;
        float val = pw[qrow * PDP + cl];
        int cidx = m0 + sub * 16 + cl;
        if (val > tv[K_ - 1]) {             // threshold gate
          tv[K_ - 1] = val;
          ti[K_ - 1] = cidx;
#pragma unroll
          for (int j = K_ - 1; j > 0; --j) {  // one guarded bubble pass
            float a0 = tv[j - 1], a1 = tv[j];
            int b0 = ti[j - 1], b1 = ti[j];
            bool sw = a1 > a0;
            tv[j - 1] = sw ? a1 : a0;
            tv[j] = sw ? a0 : a1;
            ti[j - 1] = sw ? b1 : b0;
            ti[j] = sw ? b0 : b1;
          }
        }
      }
    }
  }

  // merge the two interleaved half-lists of each query (same-wave LDS order)
  float* mvw = mv + wave * (16 * 2 * K_);
  int* miw = mi + wave * (16 * 2 * K_);
#pragma unroll
  for (int j = 0; j < K_; ++j) {
    mvw[(qrow * 2 + half) * K_ + j] = tv[j];
    miw[(qrow * 2 + half) * K_ + j] = ti[j];
  }
  if (lane < 16) {
    const float* v0p = mvw + (lane * 2 + 0) * K_;
    const float* v1p = mvw + (lane * 2 + 1) * K_;
    const int* i0p = miw + (lane * 2 + 0) * K_;
    const int* i1p = miw + (lane * 2 + 1) * K_;
    int p0 = 0, p1 = 0;
    int* op = idxOut + ((size_t)b * N_ + q0 + lane) * K_;
    for (int j = 0; j < K_; ++j) {          // p0,p1 <= j <= 19: in bounds
      float v0 = v0p[p0], v1 = v1p[p1];
      int i0 = i0p[p0], i1 = i1p[p1];
      bool t0 = (v0 > v1) || (v0 == v1 && i0 < i1);
      op[j] = t0 ? i0 : i1;
      p0 += t0 ? 1 : 0;
      p1 += t0 ? 0 : 1;
    }
  }
}

// ------- Phase 2: gather + edge GEMM + BN + LeakyReLU + max over k -------
#define WP 80   // LDS pitch for W tiles (lane-half bank-conflict free)

__global__ __launch_bounds__(128) void edge_kernel(
    const float* __restrict__ x, const float* __restrict__ W,
    const float* __restrict__ gamma, const float* __restrict__ beta,
    const float* __restrict__ rmean, const float* __restrict__ rvar,
    const int* __restrict__ knn, float* __restrict__ out) {
  __shared__ float W1T[C_ * WP];     // [c][o] = W[o][c]
  __shared__ float WdT[C_ * WP];     // [c][o] = W[o][64+c] - W[o][c]
  __shared__ float tBuf[16 * O_];    // t[p][o] = x_p . (W2-W1)[o]
  __shared__ float obuf[16 * O_];    // per-point running max
  __shared__ float rbuf[4 * 16 * O_];// per-wave per-row activated values
  __shared__ float invS[O_], shS[O_];

  const int tid = threadIdx.x;
  const int wave = tid >> 5;
  const int lane = tid & 31;
  const int lh = lane & 15;
  const int hif = (lane >> 4) & 1;
  const int b = blockIdx.y;
  const int n0 = blockIdx.x * 16;
  const float* xb = x + (size_t)b * C_ * N_;

  // stage W1 / (W2-W1) transposed, BN constants, init output max
  for (int e = tid; e < C_ * O_; e += 128) {
    int c = e >> 6, o = e & 63;
    float w1 = W[o * (2 * C_) + c];
    float w2 = W[o * (2 * C_) + C_ + c];
    W1T[c * WP + o] = w1;
    WdT[c * WP + o] = w2 - w1;
  }
  if (tid < O_) {
    float inv = gamma[tid] * rsqrtf(rvar[tid] + EPS_);
    invS[tid] = inv;
    shS[tid] = beta[tid] - rmean[tid] * inv;
  }
  for (int e = tid; e < 16 * O_; e += 128) obuf[e] = -3.0e38f;
  __syncthreads();

  // t[p][o] for the 16 workgroup points; wave w computes O-tile ot = w
  {
    const int ot = wave;
    v8f tacc = {0.f, 0.f, 0.f, 0.f, 0.f, 0.f, 0.f, 0.f};
#pragma unroll
    for (int kc = 0; kc < 16; ++kc) {
      int c0 = 4 * kc + 2 * hif;
      v2f a, bf;
      a[0] = xb[(size_t)c0 * N_ + n0 + lh];
      a[1] = xb[(size_t)(c0 + 1) * N_ + n0 + lh];
      bf[0] = WdT[c0 * WP + ot * 16 + lh];
      bf[1] = WdT[(c0 + 1) * WP + ot * 16 + lh];
      tacc = wmma4(a, bf, tacc);
    }
#pragma unroll
    for (int v = 0; v < 8; ++v) {
      int M = v + 8 * hif;
      tBuf[M * O_ + ot * 16 + lh] = tacc[v];
    }
  }
  __syncthreads();

  float inv4[4], sh4[4];
#pragma unroll
  for (int ot = 0; ot < 4; ++ot) {
    inv4[ot] = invS[ot * 16 + lh];
    sh4[ot] = shS[ot * 16 + lh];
  }

  // wave handles 4 points -> 80 (n,k) rows -> exactly 5 WMMA row-tiles
  const int* kb = knn + ((size_t)b * N_ + n0 + wave * 4) * K_;
  float* rb = rbuf + wave * (16 * O_);

#pragma unroll 1
  for (int rt = 0; rt < 5; ++rt) {
    int rowIdx = rt * 16 + lh;
    int pl = rowIdx / K_;
    int kk = rowIdx - pl * K_;
    int m = kb[pl * K_ + kk];               // neighbor point index

    v8f acc[4];
#pragma unroll
    for (int ot = 0; ot < 4; ++ot) acc[ot] = (v8f){0.f,0.f,0.f,0.f,0.f,0.f,0.f,0.f};

#pragma unroll
    for (int kc = 0; kc < 16; ++kc) {
      int c0 = 4 * kc + 2 * hif;
      v2f a;
      a[0] = xb[(size_t)c0 * N_ + m];       // gathered neighbor feature
      a[1] = xb[(size_t)(c0 + 1) * N_ + m];
#pragma unroll
      for (int ot = 0; ot < 4; ++ot) {
        v2f bf;
        bf[0] = W1T[c0 * WP + ot * 16 + lh];
        bf[1] = W1T[(c0 + 1) * WP + ot * 16 + lh];
        acc[ot] = wmma4(a, bf, acc[ot]);
      }
    }

    // finalize: + t, BN affine, LeakyReLU; stash per-row (race-free)
#pragma unroll
    for (int v = 0; v < 8; ++v) {
      int M = v + 8 * hif;
      int ri = rt * 16 + M;
      int pg = ri / K_;
#pragma unroll
      for (int ot = 0; ot < 4; ++ot) {
        int o = ot * 16 + lh;
        float val = acc[ot][v] + tBuf[(wave * 4 + pg) * O_ + o];
        val = fmaf(val, inv4[ot], sh4[ot]);
        val = (val >= 0.f) ? val : SLOPE_ * val;
        rb[M * O_ + o] = val;
      }
    }
    // reduce this tile's rows into per-point max (same-wave LDS in order)
#pragma unroll
    for (int pl2 = 0; pl2 < 4; ++pl2) {
      int lo = pl2 * K_ - rt * 16;
      int hi2 = lo + K_;
      lo = lo < 0 ? 0 : lo;
      hi2 = hi2 > 16 ? 16 : hi2;
      if (lo < hi2) {
        float m1 = -3.0e38f, m2 = -3.0e38f;
        for (int r = lo; r < hi2; ++r) {
          m1 = fmaxf(m1, rb[r * O_ + lane]);
          m2 = fmaxf(m2, rb[r * O_ + lane + 32]);
        }
        int adr = (wave * 4 + pl2) * O_;
        obuf[adr + lane] = fmaxf(obuf[adr + lane], m1);
        obuf[adr + lane + 32] = fmaxf(obuf[adr + lane + 32], m2);
      }
    }
  }
  __syncthreads();
  // out is (B, O, N)
  for (int e = tid; e < 16 * O_; e += 128) {
    int p = e & 15, o = e >> 4;
    out[((size_t)b * O_ + o) * N_ + n0 + p] = obuf[p * O_ + o];
  }
}

extern "C" void kernel_launch(void* const* d_in, const int* in_sizes, int n_in,
                              void* d_out, int out_size, void* d_ws, size_t ws_size,
                              hipStream_t stream) {
  (void)in_sizes; (void)n_in; (void)out_size; (void)ws_size;
  const float* x = (const float*)d_in[0];
  const float* W = (const float*)d_in[1];
  const float* gamma = (const float*)d_in[2];
  const float* beta = (const float*)d_in[3];
  const float* rmean = (const float*)d_in[4];
  const float* rvar = (const float*)d_in[5];
  float* out = (float*)d_out;

  float* xx = (float*)d_ws;                                         // B*N f32
  int* idx = (int*)((char*)d_ws + (size_t)B_ * N_ * sizeof(float)); // B*N*K i32

  sumsq_kernel<<<dim3((B_ * N_) / 256), 256, 0, stream>>>(x, xx);
  knn_kernel<<<dim3(N_ / QT, B_), 128, 0, stream>>>(x, xx, idx);
  edge_kernel<<<dim3(N_ / 16, B_), 128, 0, stream>>>(x, W, gamma, beta, rmean,
                                                     rvar, idx, out);
}